// ZCANormPIunstable_16106127360518
// MI455X (gfx1250) — compile-verified
//
#include <hip/hip_runtime.h>
#include <hip/hip_bf16.h>

typedef __attribute__((ext_vector_type(2))) float v2f;
typedef __attribute__((ext_vector_type(8))) float v8f;

#define GROUPS   8
#define L        64
#define CCH      512          // channels
#define HW       3136         // 56*56
#define NB       32           // batch
#define MTOT     100352       // NB*HW
#define BM       128          // columns per tile
#define CHUNKS   (MTOT / BM)  // 784
#define EPSV     1e-4f
#define PAD      129          // BM+1, LDS row stride (conflict-free)

// ---------------------------------------------------------------------------
// Kernel 1: raw second-moment S[g] += x_tile * x_tile^T  (WMMA f32 16x16x4)
//           + per-channel sums for the mean.
// ---------------------------------------------------------------------------
__global__ void __launch_bounds__(256) zca_stats_kernel(
    const float* __restrict__ x, float* __restrict__ S, float* __restrict__ chsum)
{
  __shared__ float xt[64 * PAD];
  __shared__ float part[256];

  const int g = blockIdx.y;
  const int chunk = blockIdx.x;
  const int t = threadIdx.x;

  // ---- stage 64 x 128 tile (channels x m-columns), coalesced over columns
  {
    const int col = t & 127;
    const int half = t >> 7;
    const int m = chunk * BM + col;
    const int n = m / HW;
    const int hw = m - n * HW;
    const float* base = x + ((size_t)n * CCH + (size_t)g * L) * HW + hw;
    const int c0 = half * 32;
#pragma unroll 8
    for (int c = c0; c < c0 + 32; ++c)
      xt[c * PAD + (t & 127)] = base[(size_t)c * HW];
  }
  __syncthreads();

  // ---- per-channel partial sums (4 partials per channel)
  {
    const int c = t & 63;
    const int p = t >> 6;
    const float* row = &xt[c * PAD + p * 32];
    float s = 0.f;
#pragma unroll 8
    for (int j = 0; j < 32; ++j) s += row[j];
    part[p * 64 + c] = s;
  }
  __syncthreads();
  if (t < 64) {
    float s = part[t] + part[64 + t] + part[128 + t] + part[192 + t];
    atomicAdd(&chsum[g * L + t], s);
  }

  // ---- covariance tiles: 16 tiles of 16x16 over the 64x64 block, 2 per wave
  const int lane = t & 31;
  const int wave = t >> 5;
  const int l15 = lane & 15;
  const int kk = (lane >> 4) << 1;   // A/B K-halves: VGPR0 holds K=0|2, VGPR1 K=1|3

  for (int i = 0; i < 2; ++i) {
    const int tile = wave * 2 + i;
    const int tr = tile >> 2;
    const int tc = tile & 3;
    v8f acc = {0.f, 0.f, 0.f, 0.f, 0.f, 0.f, 0.f, 0.f};
    const float* ar = &xt[(tr * 16 + l15) * PAD];   // A: rows = channels
    const float* br = &xt[(tc * 16 + l15) * PAD];   // B: cols = channels (xg^T)
#pragma unroll 8
    for (int k = 0; k < BM; k += 4) {
      v2f a; a[0] = ar[k + kk]; a[1] = ar[k + kk + 1];
      v2f b; b[0] = br[k + kk]; b[1] = br[k + kk + 1];
      acc = __builtin_amdgcn_wmma_f32_16x16x4_f32(
          false, a, false, b, (short)0, acc, false, false);
    }
    const int colg = tc * 16 + l15;
    const int rb = tr * 16 + ((lane >= 16) ? 8 : 0);
    float* dst = &S[(size_t)g * 4096 + (size_t)rb * 64 + colg];
#pragma unroll
    for (int r = 0; r < 8; ++r) atomicAdd(dst + r * 64, acc[r]);
  }
}

// ---------------------------------------------------------------------------
// Kernel 2: finalize cov = S/M - mu mu^T + eps I, then sequential power
// iteration + deflation, exactly mirroring the reference semantics.
// One block (64 threads, 2 waves) per group; A/subspace live in LDS.
// ---------------------------------------------------------------------------
__device__ __forceinline__ float block_reduce64(float v, float* red)
{
  const int tid = threadIdx.x;
  __syncthreads();               // protect red[] from previous use
  red[tid] = v;
  __syncthreads();
  if (tid < 32) {
    float s = red[tid] + red[tid + 32];
#pragma unroll
    for (int off = 16; off > 0; off >>= 1) s += __shfl_down(s, off, 32);
    if (tid == 0) red[0] = s;
  }
  __syncthreads();
  return red[0];
}

__global__ void __launch_bounds__(64) zca_power_iter_kernel(
    const float* __restrict__ S, const float* __restrict__ chsum,
    const float* __restrict__ vinit, float* __restrict__ subs)
{
  __shared__ float A[64 * 65];
  __shared__ float Sub[64 * 65];
  __shared__ float v[64];
  __shared__ float av[64];
  __shared__ float red[64];
  __shared__ float mu[64];

  const int g = blockIdx.x;
  const int tid = threadIdx.x;
  const float invM = 1.0f / (float)MTOT;

  mu[tid] = chsum[g * L + tid] * invM;
  __syncthreads();
  const float mui = mu[tid];
  for (int j = 0; j < 64; ++j) {
    A[tid * 65 + j] = S[(size_t)g * 4096 + tid * 64 + j] * invM - mui * mu[j]
                    + ((tid == j) ? EPSV : 0.f);
    Sub[tid * 65 + j] = 0.f;
  }

  float lam_prev = 0.f;
  bool stopped = false;

  for (int je = 0; je < 64; ++je) {
    // v = l2n(v_init[g][je])
    const float v0 = vinit[(size_t)g * 4096 + je * 64 + tid];
    float ss = block_reduce64(v0 * v0, red);
    float inv = 1.0f / (sqrtf(ss) + 1e-12f);
    v[tid] = v0 * inv;
    __syncthreads();

    // 19 power iterations: v = l2n(A @ v)
    for (int it = 0; it < 19; ++it) {
      float a = 0.f;
#pragma unroll 8
      for (int k = 0; k < 64; ++k) a += A[tid * 65 + k] * v[k];
      float s2 = block_reduce64(a * a, red);
      float iv = 1.0f / (sqrtf(s2) + 1e-12f);
      v[tid] = a * iv;
      __syncthreads();
    }

    // Rayleigh quotient
    float a = 0.f;
#pragma unroll 8
    for (int k = 0; k < 64; ++k) a += A[tid * 65 + k] * v[k];
    av[tid] = a;
    const float vt = v[tid];
    float vAv = block_reduce64(vt * a, red);
    float vv  = block_reduce64(vt * vt, red);
    const float lam = vAv / vv;

    const bool brk = (je > 0) && ((lam_prev < lam) || (lam < EPSV));
    const bool active = (!stopped) && (!brk);
    if (active) {
      const float rs = rsqrtf(lam);
      const float vi = v[tid];
      const float ai = av[tid];
#pragma unroll 4
      for (int k = 0; k < 64; ++k) {
        const float vk = v[k];
        Sub[tid * 65 + k] += rs * vi * vk;   // subspace += rsqrt(lam) v v^T
        A[tid * 65 + k]   -= ai * vk;        // A -= (A v) v^T  (= A - A vv^T)
      }
      lam_prev = lam;
    }
    stopped = stopped || brk;
    __syncthreads();
  }

  for (int k = 0; k < 64; ++k)
    subs[(size_t)g * 4096 + tid * 64 + k] = Sub[tid * 65 + k];
}

// ---------------------------------------------------------------------------
// Kernel 3: y = subspace @ (x - mu), * weight + bias. WMMA f32 16x16x4,
// results staged through LDS for coalesced global stores.
// ---------------------------------------------------------------------------
__global__ void __launch_bounds__(256) zca_apply_kernel(
    const float* __restrict__ x, const float* __restrict__ subs,
    const float* __restrict__ chsum, const float* __restrict__ weight,
    const float* __restrict__ bias, float* __restrict__ out)
{
  __shared__ float xt[64 * PAD];
  __shared__ float Wl[64 * 65];
  __shared__ float mu[64], wl[64], bl[64];

  const int g = blockIdx.y;
  const int chunk = blockIdx.x;
  const int t = threadIdx.x;

  if (t < 64) {
    mu[t] = chsum[g * L + t] * (1.0f / (float)MTOT);
    wl[t] = weight[g * L + t];
    bl[t] = bias[g * L + t];
  }
  for (int idx = t; idx < 4096; idx += 256)
    Wl[(idx >> 6) * 65 + (idx & 63)] = subs[(size_t)g * 4096 + idx];
  __syncthreads();

  const int col = t & 127;
  const int half = t >> 7;
  const int m = chunk * BM + col;
  const int n = m / HW;
  const int hw = m - n * HW;
  const float* base = x + ((size_t)n * CCH + (size_t)g * L) * HW + hw;
  const int c0 = half * 32;
#pragma unroll 8
  for (int c = c0; c < c0 + 32; ++c)
    xt[c * PAD + col] = base[(size_t)c * HW] - mu[c];
  __syncthreads();

  const int lane = t & 31;
  const int wave = t >> 5;
  const int l15 = lane & 15;
  const int kk = (lane >> 4) << 1;

  v8f acc[4];
  for (int i = 0; i < 4; ++i) {
    const int tile = wave * 4 + i;
    const int tr = tile >> 3;     // 0..3 output-row tile (channels)
    const int tc = tile & 7;      // 0..7 output-col tile (m columns)
    v8f c8 = {0.f, 0.f, 0.f, 0.f, 0.f, 0.f, 0.f, 0.f};
    const float* ar = &Wl[(tr * 16 + l15) * 65];
#pragma unroll 8
    for (int k = 0; k < 64; k += 4) {
      v2f a; a[0] = ar[k + kk]; a[1] = ar[k + kk + 1];
      v2f b;
      b[0] = xt[(k + kk) * PAD + tc * 16 + l15];
      b[1] = xt[(k + kk + 1) * PAD + tc * 16 + l15];
      c8 = __builtin_amdgcn_wmma_f32_16x16x4_f32(
          false, a, false, b, (short)0, c8, false, false);
    }
    acc[i] = c8;
  }
  __syncthreads();                 // all xt reads complete before overwrite

  for (int i = 0; i < 4; ++i) {
    const int tile = wave * 4 + i;
    const int tr = tile >> 3;
    const int tc = tile & 7;
    const int rb = tr * 16 + ((lane >= 16) ? 8 : 0);
    const int cc = tc * 16 + l15;
#pragma unroll
    for (int r = 0; r < 8; ++r) xt[(rb + r) * PAD + cc] = acc[i][r];
  }
  __syncthreads();

  float* obase = out + ((size_t)n * CCH + (size_t)g * L) * HW + hw;
#pragma unroll 8
  for (int c = c0; c < c0 + 32; ++c)
    obase[(size_t)c * HW] = xt[c * PAD + col] * wl[c] + bl[c];
}

// ---------------------------------------------------------------------------
extern "C" void kernel_launch(void* const* d_in, const int* in_sizes, int n_in,
                              void* d_out, int out_size, void* d_ws, size_t ws_size,
                              hipStream_t stream)
{
  const float* x      = (const float*)d_in[0];
  const float* v_init = (const float*)d_in[1];
  const float* weight = (const float*)d_in[2];
  const float* bias   = (const float*)d_in[3];
  float* out = (float*)d_out;

  // workspace layout (floats): S[8*64*64] | chsum[512] | subspace[8*64*64]
  float* S     = (float*)d_ws;
  float* chsum = S + GROUPS * L * L;
  float* subs  = chsum + CCH;

  hipMemsetAsync(d_ws, 0, (size_t)(GROUPS * L * L + CCH) * sizeof(float), stream);

  dim3 grid(CHUNKS, GROUPS);
  zca_stats_kernel<<<grid, 256, 0, stream>>>(x, S, chsum);
  zca_power_iter_kernel<<<GROUPS, 64, 0, stream>>>(S, chsum, v_init, subs);
  zca_apply_kernel<<<grid, 256, 0, stream>>>(x, subs, chsum, weight, bias, out);
}